// SEBlock_28741921145092
// MI455X (gfx1250) — compile-verified
//
#include <hip/hip_runtime.h>

typedef __attribute__((ext_vector_type(2))) float v2f;
typedef __attribute__((ext_vector_type(4))) float v4f;
typedef __attribute__((ext_vector_type(8))) float v8f;

#define EPS_BN 1e-3f

constexpr int Bsz = 32;     // batch
constexpr int C   = 512;    // channels
constexpr int HC  = 128;    // hidden channels
constexpr int HW  = 4096;   // 64*64

// ---------------------------------------------------------------------------
// Kernel A: squeeze  s[b,c] = mean(x[b,c,:,:])
// One 256-thread block per (b,c) plane (16 KB). Non-temporal B128 loads: x is
// 256 MB (> 192 MB L2) and gets re-streamed by kernel C, so don't pollute L2.
// ---------------------------------------------------------------------------
__global__ __launch_bounds__(256) void se_squeeze(const float* __restrict__ x,
                                                  float* __restrict__ s) {
    __shared__ float red[256];
    const int p = blockIdx.x;  // plane = b*C + c
    const v4f* xp = reinterpret_cast<const v4f*>(x) + (size_t)p * (HW / 4);
    const int t = threadIdx.x;
    float sum = 0.f;
#pragma unroll
    for (int j = 0; j < 4; ++j) {
        v4f v = __builtin_nontemporal_load(&xp[t + j * 256]);
        sum += v.x + v.y + v.z + v.w;
    }
    red[t] = sum;
    __syncthreads();
    for (int off = 128; off > 0; off >>= 1) {
        if (t < off) red[t] += red[t + off];
        __syncthreads();
    }
    if (t == 0) s[p] = red[0] * (1.0f / (float)HW);
}

// ---------------------------------------------------------------------------
// Kernel B: excitation, single workgroup (512 threads = 16 waves), fp32 WMMA.
//   h = s @ w1^T  -> BN1 -> ReLU -> LDS -> e = h @ w2^T -> f = e * scale2
// Fragment layouts per CDNA5 ISA 7.12.2 (wave32):
//   A 16x4 f32 : 2 VGPRs; lane<16 holds K={0,1}, lane>=16 holds K={2,3}, M=lane%16
//   B 4x16 f32 : 2 VGPRs; same K split, N=lane%16
//   C/D 16x16  : 8 VGPRs; VGPR v holds M=v (lanes 0-15) / M=v+8 (lanes 16-31)
// ---------------------------------------------------------------------------
__global__ __launch_bounds__(512) void se_excite(
    const float* __restrict__ s, const float* __restrict__ w1,
    const float* __restrict__ g1, const float* __restrict__ b1,
    const float* __restrict__ m1, const float* __restrict__ v1,
    const float* __restrict__ w2, const float* __restrict__ g2,
    const float* __restrict__ v2, float* __restrict__ f) {
    __shared__ float hbuf[Bsz * HC];  // 16 KB

    const int tid  = threadIdx.x;
    const int wave = tid >> 5;
    const int lane = tid & 31;
    const int half = lane >> 4;   // 0: lanes 0-15, 1: lanes 16-31
    const int l16  = lane & 15;
    const int k0   = half * 2;    // K offset within a 4-wide slab

    // ---- GEMM1: h[32x128] = s[32x512] * w1[128x512]^T, one 16x16 tile/wave
    {
        const int mi = wave >> 3;         // 0..1
        const int ni = wave & 7;          // 0..7
        const int m  = mi * 16 + l16;     // A row for this lane
        const int n  = ni * 16 + l16;     // B column (hidden channel)
        const float* Arow = s  + (size_t)m * C;
        const float* Brow = w1 + (size_t)n * C;   // B[k][n] = w1[n][k]
        v8f acc = {};
        for (int k = 0; k < C; k += 4) {
            v2f a = *reinterpret_cast<const v2f*>(Arow + k + k0);
            v2f b = *reinterpret_cast<const v2f*>(Brow + k + k0);
            acc = __builtin_amdgcn_wmma_f32_16x16x4_f32(
                false, a, false, b, (short)0, acc, false, false);
        }
        // BN1 + ReLU epilogue: this lane's output column is fixed (= n)
        const float sc = g1[n] * rsqrtf(v1[n] + EPS_BN);
        const float sh = b1[n] - m1[n] * sc;
#pragma unroll
        for (int v = 0; v < 8; ++v) {
            float hv = fmaxf(fmaf(acc[v], sc, sh), 0.f);
            const int row = mi * 16 + v + half * 8;
            hbuf[row * HC + n] = hv;
        }
    }
    __syncthreads();

    // ---- GEMM2: e[32x512] = h[32x128] * w2[512x128]^T, 4 tiles/wave
#pragma unroll
    for (int t = 0; t < 4; ++t) {
        const int tile = wave * 4 + t;    // 0..63
        const int mi = tile >> 5;         // 0..1
        const int ni = tile & 31;         // 0..31
        const int m  = mi * 16 + l16;
        const int c  = ni * 16 + l16;     // output channel for this lane
        const float* Arow = hbuf + m * HC;
        const float* Brow = w2 + (size_t)c * HC;  // B[k][c] = w2[c][k]
        v8f acc = {};
#pragma unroll 4
        for (int k = 0; k < HC; k += 4) {
            v2f a = *reinterpret_cast<const v2f*>(Arow + k + k0);
            v2f b = *reinterpret_cast<const v2f*>(Brow + k + k0);
            acc = __builtin_amdgcn_wmma_f32_16x16x4_f32(
                false, a, false, b, (short)0, acc, false, false);
        }
        const float sc2 = g2[c] * rsqrtf(v2[c] + EPS_BN);
#pragma unroll
        for (int v = 0; v < 8; ++v) {
            const int row = mi * 16 + v + half * 8;
            f[row * C + c] = acc[v] * sc2;   // f = e * scale2
        }
    }
}

// ---------------------------------------------------------------------------
// Kernel C: y = x * f[b,c] + shift2[c]   (pure HBM stream, B128 NT in/out)
// shift2 recomputed per thread from the 2 KB param vectors (L2-resident).
// ---------------------------------------------------------------------------
__global__ __launch_bounds__(256) void se_scale(
    const float* __restrict__ x, const float* __restrict__ f,
    const float* __restrict__ g2, const float* __restrict__ b2,
    const float* __restrict__ m2, const float* __restrict__ v2,
    float* __restrict__ y) {
    const size_t i4 = (size_t)blockIdx.x * 256 + threadIdx.x;
    const int bc = (int)(i4 >> 10);        // 1024 x float4 per (b,c) plane
    const int c  = bc & (C - 1);
    const float a   = f[bc];               // e * scale2
    const float sc2 = g2[c] * rsqrtf(v2[c] + EPS_BN);
    const float sh  = b2[c] - m2[c] * sc2;
    const v4f* xp = reinterpret_cast<const v4f*>(x);
    v4f* yp = reinterpret_cast<v4f*>(y);
    v4f xv = __builtin_nontemporal_load(&xp[i4]);
    v4f r;
    r.x = fmaf(xv.x, a, sh);
    r.y = fmaf(xv.y, a, sh);
    r.z = fmaf(xv.z, a, sh);
    r.w = fmaf(xv.w, a, sh);
    __builtin_nontemporal_store(r, &yp[i4]);
}

extern "C" void kernel_launch(void* const* d_in, const int* in_sizes, int n_in,
                              void* d_out, int out_size, void* d_ws, size_t ws_size,
                              hipStream_t stream) {
    const float* x  = (const float*)d_in[0];
    const float* w1 = (const float*)d_in[1];
    const float* g1 = (const float*)d_in[2];
    const float* b1 = (const float*)d_in[3];
    const float* m1 = (const float*)d_in[4];
    const float* v1 = (const float*)d_in[5];
    const float* w2 = (const float*)d_in[6];
    const float* g2 = (const float*)d_in[7];
    const float* b2 = (const float*)d_in[8];
    const float* m2 = (const float*)d_in[9];
    const float* v2 = (const float*)d_in[10];
    float* out = (float*)d_out;

    float* s = (float*)d_ws;            // [32*512]
    float* f = s + Bsz * C;             // [32*512]

    se_squeeze<<<Bsz * C, 256, 0, stream>>>(x, s);
    se_excite<<<1, 512, 0, stream>>>(s, w1, g1, b1, m1, v1, w2, g2, v2, f);
    const int n4 = Bsz * C * HW / 4;    // 16,777,216 float4
    se_scale<<<n4 / 256, 256, 0, stream>>>(x, f, g2, b2, m2, v2, out);
}